// CRF_2327872274919
// MI455X (gfx1250) — compile-verified
//
#include <hip/hip_runtime.h>
#include <stdint.h>

#define T 32
#define BATCH 1024
#define SEQ 512
#define PREF 8
#define RING 16

// Broadcast lane l's value of x to all lanes (wave32, uniform l).
__device__ __forceinline__ float lane_bcast(float x, int l) {
  return __uint_as_float((unsigned)__builtin_amdgcn_readlane((int)__float_as_uint(x), l));
}

// CDNA5 async global->LDS load: each lane moves 4B from its global address to
// its LDS byte offset. Tracked by ASYNCcnt (CDNA5 ISA).
__device__ __forceinline__ void async_feat_load(unsigned lds_off, const float* g) {
  asm volatile("global_load_async_to_lds_b32 %0, %1, off"
               :: "v"(lds_off), "v"(g) : "memory");
}

// 8-long argmax chain over prev = base..base+7.
// Ascending scan with strict '>' keeps the smallest index on ties (jnp.argmax).
#define CHAIN8(BEST, BP, BASE)                                                 \
  float BEST = lane_bcast(s, (BASE)) + tcol[(BASE)];                           \
  int BP = (BASE);                                                             \
  _Pragma("unroll")                                                            \
  for (int p = (BASE) + 1; p < (BASE) + 8; ++p) {                              \
    float c = lane_bcast(s, p) + tcol[p];                                      \
    if (c > BEST) { BEST = c; BP = p; }                                        \
  }

// One Viterbi step: s = per-lane score (lane == cur tag), tcol[p] = trans[p][lane].
// 4 independent chains for VOPD dual-issue; merges use strict '>' from the
// higher-index half so the overall winner is the smallest argmax index.
#define VSTEP(ii, FEAT)                                                        \
  {                                                                            \
    CHAIN8(b0, p0, 0)                                                          \
    CHAIN8(b1, p1, 8)                                                          \
    CHAIN8(b2, p2, 16)                                                         \
    CHAIN8(b3, p3, 24)                                                         \
    if (b1 > b0) { b0 = b1; p0 = p1; }                                         \
    if (b3 > b2) { b2 = b3; p2 = p3; }                                         \
    if (b2 > b0) { b0 = b2; p0 = p2; }                                         \
    s = b0 + (FEAT);                                                           \
    bpWord |= ((unsigned)p0) << (((ii) & 3) * 8);                              \
    if (((ii) & 3) == 3) { bpW[((ii) >> 2) * T + lane] = bpWord; bpWord = 0; } \
  }

__global__ __launch_bounds__(32) void crf_viterbi_kernel(
    const float* __restrict__ feats,       // (B, S, 32)
    const float* __restrict__ trans,       // (32, 32) [prev][cur]
    const float* __restrict__ startT,      // (32,)
    const float* __restrict__ stopT,       // (32,)
    int* __restrict__ out)                 // (B, S) int32
{
  __shared__ float ring[RING * T];             // 2 KB feat staging ring
  __shared__ unsigned bpW[(SEQ / 4) * T];      // 16 KB packed backpointers
  __shared__ unsigned char path[SEQ];          // 512 B decoded path

  const int lane = threadIdx.x;
  const int b = blockIdx.x;
  const float* fb = feats + (size_t)b * SEQ * T;

  // LDS byte offset of the ring (low 32 bits of the flat LDS-aperture addr).
  unsigned ringBase = (unsigned)(size_t)(void*)ring;

  // Lane `cur` caches transition column trans[0..31][cur] in 32 VGPRs.
  float tcol[T];
#pragma unroll
  for (int p = 0; p < T; ++p) tcol[p] = trans[p * T + lane];

  // Prologue: put PREF feat rows (128 B each) in flight.
#pragma unroll
  for (int j = 0; j < PREF; ++j)
    async_feat_load(ringBase + (unsigned)j * (T * 4) + lane * 4, fb + j * T + lane);

  asm volatile("s_wait_asynccnt 7" ::: "memory");   // row 0 complete (in-order)
  float s = ring[lane] + startT[lane];              // pre = feats[:,0] + start
  asm volatile("s_wait_dscnt 0" ::: "memory");
  async_feat_load(ringBase + (PREF & (RING - 1)) * (T * 4) + lane * 4,
                  fb + PREF * T + lane);

  unsigned bpWord = 0;

  // Steady state (branch-free): before consuming row i, issued = i+8 loads and
  // we need i+1 complete -> s_wait_asynccnt 7 guarantees row i is in LDS.
  // i+PREF <= 511 here, so the prefetch is unconditional.
  for (int i = 1; i <= SEQ - PREF - 1; ++i) {
    asm volatile("s_wait_asynccnt 7" ::: "memory");
    float feat = ring[(i & (RING - 1)) * T + lane];
    VSTEP(i, feat);
    int nf = i + PREF;
    // The ds_load of the recycled slot finished >=8 iterations ago; the wait
    // makes the LDS-read -> async-overwrite ordering architectural.
    asm volatile("s_wait_dscnt 0" ::: "memory");
    async_feat_load(ringBase + (nf & (RING - 1)) * (T * 4) + lane * 4,
                    fb + nf * T + lane);
  }

  // i = SEQ-PREF (504): issued = 512 total, wait<=7 -> rows 0..504 complete.
  {
    const int i = SEQ - PREF;
    asm volatile("s_wait_asynccnt 7" ::: "memory");
    float feat = ring[(i & (RING - 1)) * T + lane];
    VSTEP(i, feat);
  }

  // Tail: everything issued; drain once, then no more waits.
  asm volatile("s_wait_asynccnt 0" ::: "memory");
#pragma unroll
  for (int i = SEQ - PREF + 1; i < SEQ; ++i) {
    float feat = ring[(i & (RING - 1)) * T + lane];
    VSTEP(i, feat);
  }

  // final = pre + stop ; last = argmax over tags (smallest index on ties)
  float fin = s + stopT[lane];
  float bestF = lane_bcast(fin, 0);
  int last = 0;
#pragma unroll
  for (int t = 1; t < T; ++t) {
    float f = lane_bcast(fin, t);
    if (f > bestF) { bestF = f; last = t; }
  }

  // Backtrack entirely in LDS (dependent chain of ds loads, ~25cy each,
  // vs ~hundreds of ns per dependent HBM load if bp lived in global).
  if (lane == 0) {
    int tag = last;
    path[SEQ - 1] = (unsigned char)tag;
    for (int j = SEQ - 1; j >= 1; --j) {
      unsigned w = bpW[(j >> 2) * T + tag];
      tag = (int)((w >> ((j & 3) * 8)) & 31u);
      path[j - 1] = (unsigned char)tag;
    }
  }
  __syncthreads();

  // Coalesced int32 output, non-temporal (write-once, keep out of L2).
  int* ob = out + (size_t)b * SEQ;
#pragma unroll
  for (int k = 0; k < SEQ / T; ++k)
    __builtin_nontemporal_store((int)path[k * T + lane], ob + k * T + lane);
}

extern "C" void kernel_launch(void* const* d_in, const int* in_sizes, int n_in,
                              void* d_out, int out_size, void* d_ws, size_t ws_size,
                              hipStream_t stream) {
  const float* feats  = (const float*)d_in[0];
  // d_in[1] = tags (unused: decode only)
  const float* trans  = (const float*)d_in[2];
  const float* startT = (const float*)d_in[3];
  const float* stopT  = (const float*)d_in[4];
  int* out = (int*)d_out;

  (void)in_sizes; (void)n_in; (void)out_size; (void)d_ws; (void)ws_size;

  dim3 grid(BATCH), block(32);
  hipLaunchKernelGGL(crf_viterbi_kernel, grid, block, 0, stream,
                     feats, trans, startT, stopT, out);
}